// SeFT_74646531605091
// MI455X (gfx1250) — compile-verified
//
#include <hip/hip_runtime.h>
#include <float.h>
#include <math.h>

// ---------------- problem constants (from reference) ----------------
#define NBATCH 4
#define NKEY   2048
#define NPTS   4096
#define CIN    61
#define CH     64      // IN = DIM + C_IN
#define HID1   128
#define HID2   128
#define OUTC   64
#define NHEADS 4
#define NB     32      // neighbors
#define BNEPS  1e-5f

#define NKEYS_TOT (NBATCH*NKEY)   // 8192
#define NROWS_TOT (NKEYS_TOT*NB)  // 262144

typedef __attribute__((ext_vector_type(2))) float v2f;
typedef __attribute__((ext_vector_type(8))) float v8f;

static __device__ __forceinline__ v8f wmma_f32(v2f a, v2f b, v8f c) {
  // V_WMMA_F32_16X16X4_F32 : D = A(16x4) x B(4x16) + C(16x16), full fp32
  return __builtin_amdgcn_wmma_f32_16x16x4_f32(false, a, false, b,
                                               (short)0, c, false, false);
}

static __device__ __forceinline__ v8f zero8() {
  v8f z = {0.f, 0.f, 0.f, 0.f, 0.f, 0.f, 0.f, 0.f};
  return z;
}

// =====================================================================
// Kernel 1: per-key kNN (32 smallest squared distances), build comb rows
// [kid*32+r][64] = [rel(3) | feat(61)], and per-key BN partial sums.
// Neighbor ORDER is irrelevant: every downstream op (MLP, sum, softmax,
// weighted reduce, batch stats) is symmetric over the neighbor axis.
// =====================================================================
__global__ __launch_bounds__(256) void knn_comb_kernel(
    const float* __restrict__ keys, const float* __restrict__ points,
    const float* __restrict__ feats, float* __restrict__ comb,
    float* __restrict__ psum, float* __restrict__ psq) {
  const int kid = blockIdx.x;           // 0..8191 flattened (b,key)
  const int b   = kid / NKEY;
  const int t   = threadIdx.x;
  const int lane = t & 31, wid = t >> 5;

  __shared__ float dist[NPTS];          // 16 KB
  __shared__ int   sel[NB];
  __shared__ float redv[8];
  __shared__ int   redi[8];
  __shared__ float combs[NB * CH];      // 8 KB

  const float kx = keys[kid*3 + 0];
  const float ky = keys[kid*3 + 1];
  const float kz = keys[kid*3 + 2];
  const float* __restrict__ pb = points + (size_t)b * NPTS * 3;

  // squared distances (sqrt is monotone -> same top-k set)
  for (int it = 0; it < NPTS/256; ++it) {
    int j = it*256 + t;
    float dx = pb[j*3+0] - kx;
    float dy = pb[j*3+1] - ky;
    float dz = pb[j*3+2] - kz;
    dist[j] = dx*dx + dy*dy + dz*dz;
  }
  __syncthreads();

  // 32x iterative argmin (deterministic tie-break on lower index)
  for (int s = 0; s < NB; ++s) {
    float v = FLT_MAX; int bi = 0x7fffffff;
    for (int it = 0; it < NPTS/256; ++it) {
      int j = it*256 + t;
      float d = dist[j];
      if (d < v) { v = d; bi = j; }
    }
    for (int off = 16; off >= 1; off >>= 1) {
      float ov = __shfl_down(v, off);
      int   oi = __shfl_down(bi, off);
      if (ov < v || (ov == v && oi < bi)) { v = ov; bi = oi; }
    }
    if (lane == 0) { redv[wid] = v; redi[wid] = bi; }
    __syncthreads();
    if (t == 0) {
      float bv = redv[0]; int bj = redi[0];
      for (int w = 1; w < 8; ++w)
        if (redv[w] < bv || (redv[w] == bv && redi[w] < bj)) { bv = redv[w]; bj = redi[w]; }
      sel[s] = bj;
      dist[bj] = FLT_MAX;   // exclude from next rounds
    }
    __syncthreads();
  }

  // build comb rows: [nrel(3) | nfeat(61)]
  const float* __restrict__ fb = feats + (size_t)b * NPTS * CIN;
  for (int e = 0; e < (NB*CH)/256; ++e) {   // 8 iters
    int flat = e*256 + t;
    int row = flat >> 6, ch = flat & 63;
    int j = sel[row];
    float val;
    if      (ch == 0) val = pb[j*3+0] - kx;
    else if (ch == 1) val = pb[j*3+1] - ky;
    else if (ch == 2) val = pb[j*3+2] - kz;
    else              val = fb[j*CIN + (ch-3)];
    combs[flat] = val;
    comb[(size_t)kid * (NB*CH) + flat] = val;
  }
  __syncthreads();

  // per-key per-channel partial sums for BatchNorm variance
  if (t < CH) {
    float s = 0.f, s2 = 0.f;
    for (int r = 0; r < NB; ++r) { float x = combs[r*CH + t]; s += x; s2 += x*x; }
    psum[kid*CH + t] = s;
    psq [kid*CH + t] = s2;
  }
}

// =====================================================================
// Kernel 2: reduce BN partials -> scale[c] = gamma*rsqrt(var+eps), then
// fold attention into We_eff[c][h] = scale[c] * sum_o att_w[c,h*64+o]*att_q[h,o].
// (mean/beta/att_b/set_feat branch cancel under softmax shift-invariance.)
// =====================================================================
__global__ __launch_bounds__(256) void stats_fold_kernel(
    const float* __restrict__ psum, const float* __restrict__ psq,
    const float* __restrict__ att_w, const float* __restrict__ att_q,
    const float* __restrict__ bn_gamma, float* __restrict__ Wef) {
  __shared__ float ls[4][CH], lq[4][CH], scale[CH];
  const int t = threadIdx.x;
  const int part = t >> 6, ch = t & 63;

  float s = 0.f, q = 0.f;
  const int chunk = NKEYS_TOT / 4;
  for (int i = part*chunk; i < (part+1)*chunk; ++i) {
    s += psum[i*CH + ch];
    q += psq [i*CH + ch];
  }
  ls[part][ch] = s; lq[part][ch] = q;
  __syncthreads();

  if (t < CH) {
    float S = ls[0][t] + ls[1][t] + ls[2][t] + ls[3][t];
    float Q = lq[0][t] + lq[1][t] + lq[2][t] + lq[3][t];
    float inv_n = 1.0f / (float)NROWS_TOT;
    float mean = S * inv_n;
    float var  = Q * inv_n - mean*mean;
    scale[t] = bn_gamma[t] * rsqrtf(var + BNEPS);
  }
  __syncthreads();

  const int c = t >> 2, h = t & 3;     // 64 x 4
  float acc = 0.f;
  for (int o = 0; o < CH; ++o)
    acc += att_w[c*(NHEADS*CH) + h*CH + o] * att_q[h*CH + o];
  Wef[c*NHEADS + h] = scale[c] * acc;
}

// =====================================================================
// Kernel 3: fully fused per-key tile (32 rows):
//   lat = MLP(comb) via chained V_WMMA_F32_16X16X4_F32 GEMMs,
//   e   = comb @ We_eff (fp32 VALU),
//   softmax over neighbors, out[k,o,h] = sum_k lat*w.
// 2 waves / block; each wave owns a 16-row M-tile.
// f32 WMMA layouts: A(16x4): K = v + 2*(lane>>4), M = lane&15
//                   B(4x16): K = v + 2*(lane>>4), N = lane&15
//                   C/D:     M = v + 8*(lane>>4), N = lane&15
// =====================================================================
__global__ __launch_bounds__(64) void fused_mlp_attn_kernel(
    const float* __restrict__ comb, const float* __restrict__ Wef,
    const float* __restrict__ w1, const float* __restrict__ b1,
    const float* __restrict__ w2, const float* __restrict__ b2,
    const float* __restrict__ w3, const float* __restrict__ b3,
    float* __restrict__ out) {
  const int kid  = blockIdx.x;
  const int t    = threadIdx.x;
  const int lane = t & 31, wid = t >> 5;

  __shared__ float combs[NB][CH];       //  8 KB, row-major
  __shared__ float h1s[HID1][NB];       // 16 KB, col-major [col][row]
  __shared__ float h2s[HID2][NB];       // 16 KB
  __shared__ float lats[OUTC][NB];      //  8 KB, col-major
  __shared__ float Wes[CH][NHEADS];     //  1 KB
  __shared__ float es[NB][NHEADS];
  __shared__ float wsm[NB][NHEADS];
  __shared__ float smax[NHEADS], ssum[NHEADS];

  for (int i = 0; i < (NB*CH)/64; ++i) {    // 32
    int flat = i*64 + t;
    combs[flat >> 6][flat & 63] = comb[(size_t)kid * (NB*CH) + flat];
  }
  for (int i = 0; i < (CH*NHEADS)/64; ++i) { // 4
    int flat = i*64 + t;
    Wes[flat >> 2][flat & 3] = Wef[flat];
  }
  __syncthreads();

  // ---- attention logits: e[row][h] = sum_c comb*scale*W_e (fp32) ----
  for (int i = 0; i < 2; ++i) {
    int idx = t*2 + i;                  // 128 values
    int row = idx >> 2, h = idx & 3;
    float acc = 0.f;
    for (int c = 0; c < CH; ++c) acc += combs[row][c] * Wes[c][h];
    es[row][h] = acc;
  }

  const int m     = lane & 15;
  const int khalf = 2 * (lane >> 4);
  const int mrow0 = wid * 16;
  const int r0    = mrow0 + 8 * (lane >> 4);

  // ---- GEMM1: comb(32x64) @ w1(64x128), +b1, ReLU -> h1 ----
  {
    v8f acc[8];
    for (int nt = 0; nt < 8; ++nt) acc[nt] = zero8();
    for (int kt = 0; kt < CH/4; ++kt) {           // 16
      int k0 = kt*4 + khalf;
      v2f a; a.x = combs[mrow0 + m][k0]; a.y = combs[mrow0 + m][k0 + 1];
      for (int nt = 0; nt < 8; ++nt) {
        int n = nt*16 + m;
        v2f bf; bf.x = w1[k0*HID1 + n]; bf.y = w1[(k0+1)*HID1 + n];
        acc[nt] = wmma_f32(a, bf, acc[nt]);
      }
    }
    for (int nt = 0; nt < 8; ++nt) {
      int n = nt*16 + m;
      float bb = b1[n];
      for (int v = 0; v < 8; ++v)
        h1s[n][r0 + v] = fmaxf(acc[nt][v] + bb, 0.f);
    }
  }
  __syncthreads();

  // ---- GEMM2: h1(32x128) @ w2(128x128), +b2, ReLU -> h2 ----
  {
    v8f acc[8];
    for (int nt = 0; nt < 8; ++nt) acc[nt] = zero8();
    for (int kt = 0; kt < HID1/4; ++kt) {         // 32
      int k0 = kt*4 + khalf;
      v2f a; a.x = h1s[k0][mrow0 + m]; a.y = h1s[k0+1][mrow0 + m];
      for (int nt = 0; nt < 8; ++nt) {
        int n = nt*16 + m;
        v2f bf; bf.x = w2[k0*HID2 + n]; bf.y = w2[(k0+1)*HID2 + n];
        acc[nt] = wmma_f32(a, bf, acc[nt]);
      }
    }
    for (int nt = 0; nt < 8; ++nt) {
      int n = nt*16 + m;
      float bb = b2[n];
      for (int v = 0; v < 8; ++v)
        h2s[n][r0 + v] = fmaxf(acc[nt][v] + bb, 0.f);
    }
  }
  __syncthreads();

  // ---- GEMM3: h2(32x128) @ w3(128x64), +b3 -> lat ----
  {
    v8f acc[4];
    for (int nt = 0; nt < 4; ++nt) acc[nt] = zero8();
    for (int kt = 0; kt < HID2/4; ++kt) {         // 32
      int k0 = kt*4 + khalf;
      v2f a; a.x = h2s[k0][mrow0 + m]; a.y = h2s[k0+1][mrow0 + m];
      for (int nt = 0; nt < 4; ++nt) {
        int n = nt*16 + m;
        v2f bf; bf.x = w3[k0*OUTC + n]; bf.y = w3[(k0+1)*OUTC + n];
        acc[nt] = wmma_f32(a, bf, acc[nt]);
      }
    }
    for (int nt = 0; nt < 4; ++nt) {
      int n = nt*16 + m;
      float bb = b3[n];
      for (int v = 0; v < 8; ++v)
        lats[n][r0 + v] = acc[nt][v] + bb;
    }
  }
  __syncthreads();

  // ---- softmax over the 32 neighbors, per head ----
  if (t < NHEADS) {
    float mx = -FLT_MAX;
    for (int k = 0; k < NB; ++k) mx = fmaxf(mx, es[k][t]);
    float s = 0.f;
    for (int k = 0; k < NB; ++k) s += expf(es[k][t] - mx);
    smax[t] = mx; ssum[t] = s;
  }
  __syncthreads();
  for (int i = 0; i < 2; ++i) {
    int idx = t*2 + i;
    int row = idx >> 2, h = idx & 3;
    wsm[row][h] = expf(es[row][h] - smax[h]) / ssum[h];
  }
  __syncthreads();

  // ---- weighted reduce: out[kid, o*4+h] = sum_k lat[k][o] * w[k][h] ----
  for (int i = 0; i < 4; ++i) {
    int idx = i*64 + t;                 // 256 outputs
    int o = idx >> 2, h = idx & 3;
    float s = 0.f;
    for (int k = 0; k < NB; ++k) s += lats[o][k] * wsm[k][h];
    out[(size_t)kid * (OUTC*NHEADS) + idx] = s;
  }
}

// =====================================================================
extern "C" void kernel_launch(void* const* d_in, const int* in_sizes, int n_in,
                              void* d_out, int out_size, void* d_ws, size_t ws_size,
                              hipStream_t stream) {
  (void)in_sizes; (void)n_in; (void)out_size; (void)ws_size;
  const float* keys   = (const float*)d_in[0];
  const float* points = (const float*)d_in[1];
  const float* feats  = (const float*)d_in[2];
  const float* nx_w1  = (const float*)d_in[3];
  const float* nx_b1  = (const float*)d_in[4];
  const float* nx_w2  = (const float*)d_in[5];
  const float* nx_b2  = (const float*)d_in[6];
  const float* nx_w3  = (const float*)d_in[7];
  const float* nx_b3  = (const float*)d_in[8];
  // d_in[9..14] = sx_* : provably cancel under softmax shift-invariance
  const float* att_w  = (const float*)d_in[15];
  // d_in[16] = att_b : cancels
  const float* att_q  = (const float*)d_in[17];
  const float* gamma  = (const float*)d_in[18];
  // d_in[19] = bn_beta : cancels; d_in[20] = neighbors (==32, static)

  float* out  = (float*)d_out;
  float* comb = (float*)d_ws;                                 // 262144*64 f32 = 64 MB
  float* psum = comb + (size_t)NROWS_TOT * CH;                // 8192*64
  float* psq  = psum + (size_t)NKEYS_TOT * CH;                // 8192*64
  float* Wef  = psq  + (size_t)NKEYS_TOT * CH;                // 64*4

  knn_comb_kernel<<<NKEYS_TOT, 256, 0, stream>>>(keys, points, feats, comb, psum, psq);
  stats_fold_kernel<<<1, 256, 0, stream>>>(psum, psq, att_w, att_q, gamma, Wef);
  fused_mlp_attn_kernel<<<NKEYS_TOT, 64, 0, stream>>>(
      comb, Wef, nx_w1, nx_b1, nx_w2, nx_b2, nx_w3, nx_b3, out);
}